// SumLayer_65360812310793
// MI455X (gfx1250) — compile-verified
//
#include <hip/hip_runtime.h>
#include <stdint.h>

// Sum-layer forward (PyJuice-style) for MI455X / gfx1250.
//
// out[n,b] = log(clip(sum_c params[pids[n,c]] * exp(em[cids[n,c],b] - m[n,b]), 1e-10)) + m[n,b]
//   m[n,b] = max_c em[cids[n,c],b]
//
// Strategy: per-node 16KB child tile staged global->LDS with CDNA5 async
// copies (ASYNCcnt), double-buffered across nodes so gather latency overlaps
// the exp/log compute. element_mars (128MB) is L2-resident (192MB L2);
// output uses non-temporal stores to keep it resident.

#define N_NODES 65536
#define N_CHS   16
#define BATCH   256
#define NPB     16            // nodes processed per block (pipeline depth)

__global__ __launch_bounds__(256) void sum_layer_async_kernel(
    const float* __restrict__ em,      // [CH_SIZE, BATCH]
    const float* __restrict__ params,  // [N_PARAMS]
    const int*   __restrict__ nids,    // [N_NODES]
    const int*   __restrict__ cids,    // [N_NODES, N_CHS]
    const int*   __restrict__ pids,    // [N_NODES, N_CHS]
    float*       __restrict__ out)     // [N_NODES, BATCH]
{
  // double-buffered tile: 2 x (16 rows x 256 cols) floats = 32 KB of 320 KB LDS
  __shared__ float lds[2 * N_CHS * BATCH];

  const int tid   = threadIdx.x;
  const int wave  = tid >> 5;       // 8 waves per block (wave32)
  const int lane  = tid & 31;
  const int node0 = blockIdx.x * NPB;

  const float LOG2E = 1.4426950408889634f;
  const float LN2   = 0.6931471805599453f;

  // Stage one node's 16x256 f32 tile into LDS buffer `buf`.
  // 32 async-B128 chunks of 512B; each wave issues 4 (ASYNCcnt += 4).
  // chunk k: row = k/2, half-row = k%2; lane moves 16 bytes.
  auto issue_tile = [&](int n, int buf) {
#pragma unroll
    for (int j = 0; j < 4; ++j) {
      const int chunk = (wave << 2) | j;
      const int row   = chunk >> 1;
      const int half  = chunk & 1;
      const int cid   = cids[n * N_CHS + row];
      const float* g  = em + (size_t)cid * BATCH + half * 128 + lane * 4;
      const unsigned l = (unsigned)(uintptr_t)
          (&lds[buf * (N_CHS * BATCH) + row * BATCH + half * 128 + lane * 4]);
      asm volatile("global_load_async_to_lds_b128 %0, %1, off"
                   :: "v"(l), "v"(g) : "memory");
    }
  };

  issue_tile(node0, 0);  // prologue: buffer 0 in flight (ASYNCcnt = 4)

  for (int i = 0; i < NPB; ++i) {
    if (i + 1 < NPB) {
      issue_tile(node0 + i + 1, (i + 1) & 1);          // ASYNCcnt -> 8
      // async loads complete in order: <=4 left  =>  current tile landed
      asm volatile("s_wait_asynccnt 4" ::: "memory");
    } else {
      asm volatile("s_wait_asynccnt 0" ::: "memory");
    }
    __syncthreads();  // all 8 waves' chunks visible in LDS

    const int n = node0 + i;

    // per-node weights: indices uniform across lanes -> scalar loads
    float w[N_CHS];
#pragma unroll
    for (int c = 0; c < N_CHS; ++c) w[c] = params[pids[n * N_CHS + c]];

    // thread t owns batch column t; LDS reads are bank-conflict-free
    const float* lf = &lds[(i & 1) * (N_CHS * BATCH)];
    float v[N_CHS];
    float m = -__builtin_inff();
#pragma unroll
    for (int c = 0; c < N_CHS; ++c) {
      v[c] = lf[c * BATCH + tid];
      m = __builtin_fmaxf(m, v[c]);
    }

    const float mb = m * LOG2E;
    float s = 0.0f;
#pragma unroll
    for (int c = 0; c < N_CHS; ++c) {
      // exp(v - m) = exp2(v*log2e - m*log2e): v_fma + v_exp (trans co-exec)
      const float e = __builtin_amdgcn_exp2f(__builtin_fmaf(v[c], LOG2E, -mb));
      s = __builtin_fmaf(w[c], e, s);
    }
    s = __builtin_fmaxf(s, 1e-10f);
    const float r = __builtin_fmaf(__builtin_amdgcn_logf(s), LN2, m); // ln = log2*ln2

    // NT store: output (64MB) is write-once, keep element_mars hot in L2
    __builtin_nontemporal_store(r, &out[(size_t)nids[n] * BATCH + tid]);

    __syncthreads();  // protect LDS buffer reuse by next iteration's asyncs
  }
}

extern "C" void kernel_launch(void* const* d_in, const int* in_sizes, int n_in,
                              void* d_out, int out_size, void* d_ws, size_t ws_size,
                              hipStream_t stream) {
  // setup_inputs order: node_mars(unused; fully overwritten), element_mars,
  // params, nids, cids, pids
  const float* em     = (const float*)d_in[1];
  const float* params = (const float*)d_in[2];
  const int*   nids   = (const int*)d_in[3];
  const int*   cids   = (const int*)d_in[4];
  const int*   pids   = (const int*)d_in[5];
  float* out = (float*)d_out;

  dim3 grid(N_NODES / NPB);   // 4096 blocks
  dim3 block(256);            // 8 wave32 waves
  sum_layer_async_kernel<<<grid, block, 0, stream>>>(em, params, nids, cids, pids, out);
}